// GATNet_48464410968770
// MI455X (gfx1250) — compile-verified
//
#include <hip/hip_runtime.h>
#include <hip/hip_bf16.h>
#include <math.h>

typedef float v2f __attribute__((ext_vector_type(2)));
typedef float v8f __attribute__((ext_vector_type(8)));

// ---------------------------------------------------------------------------
// WMMA GEMM:  Out[M x Cout] = A[M x K] @ W[K x Cout]
// fp32 inputs -> V_WMMA_F32_16X16X4_F32 (exact fp32, no downcast).
// Block = 128 threads (4 waves). Wave w computes the 16x16 tile (m0, 16*w).
// Requires: M % 16 == 0, K % 4 == 0, Cout <= 64. W staged in LDS, padded to 64 cols.
// ---------------------------------------------------------------------------
__global__ void gat_gemm_wmma(const float* __restrict__ A,
                              const float* __restrict__ W,
                              float* __restrict__ Out,
                              int K, int Cout)
{
    __shared__ float Ws[128 * 64];
    const int total = K * 64;
    for (int i = threadIdx.x; i < total; i += blockDim.x) {
        int kk = i >> 6, nn = i & 63;
        Ws[i] = (nn < Cout) ? W[kk * Cout + nn] : 0.0f;
    }
    __syncthreads();

    const int lane = threadIdx.x & 31;
    const int wave = threadIdx.x >> 5;
    const int m0   = blockIdx.x << 4;
    const int n0   = wave << 4;
    const int row  = m0 + (lane & 15);
    const int kh   = (lane >> 4) << 1;   // 0 for lanes 0-15, 2 for lanes 16-31
    const int col  = lane & 15;

    v8f acc = {};
    const float* arow = A + (size_t)row * K;
    for (int k = 0; k < K; k += 4) {
        v2f a, b;
        // A 16x4 fp32 layout: V0 = K(0|2), V1 = K(1|3), split by lane half
        a.x = arow[k + kh];
        a.y = arow[k + kh + 1];
        // B 4x16 fp32 layout: lanes 0-15 hold K=0..1, lanes 16-31 hold K=2..3
        b.x = Ws[(k + kh) * 64 + n0 + col];
        b.y = Ws[(k + kh + 1) * 64 + n0 + col];
        acc = __builtin_amdgcn_wmma_f32_16x16x4_f32(false, a, false, b,
                                                    (short)0, acc, false, false);
    }

    // C/D layout: VGPR j -> row m0+j (lanes 0-15) or m0+8+j (lanes 16-31)
    const int orow = m0 + ((lane >> 4) << 3);
    const int ocol = n0 + col;
    if (ocol < Cout) {
#pragma unroll
        for (int j = 0; j < 8; ++j)
            Out[(size_t)(orow + j) * Cout + ocol] = acc[j];
    }
}

// ---------------------------------------------------------------------------
// Per-(node, head) attention scores:  as[n,h] = <h1[n,h,:], att_src[h,:]>
// ---------------------------------------------------------------------------
__global__ void gat_att_scores(const float* __restrict__ h,
                               const float* __restrict__ att_s,
                               const float* __restrict__ att_d,
                               float* __restrict__ as_, float* __restrict__ ad_,
                               int N, int H, int C)
{
    int t = blockIdx.x * blockDim.x + threadIdx.x;
    if (t >= N * H) return;
    int hh = t % H;
    const float* hp = h + (size_t)t * C;
    float ss = 0.f, dd = 0.f;
    for (int c = 0; c < C; ++c) {
        float v = hp[c];
        ss += v * att_s[hh * C + c];
        dd += v * att_d[hh * C + c];
    }
    as_[t] = ss; ad_[t] = dd;
}

// monotone float <-> sortable uint
__device__ __forceinline__ unsigned flip_f(float f) {
    unsigned u = __float_as_uint(f);
    return (u & 0x80000000u) ? ~u : (u | 0x80000000u);
}
__device__ __forceinline__ float unflip_f(unsigned u) {
    return __uint_as_float((u & 0x80000000u) ? (u & 0x7FFFFFFFu) : ~u);
}

__device__ __forceinline__ void edge_endpoints(const int* __restrict__ ei,
                                               int e, int E, int& s, int& d) {
    if (e < E) { s = ei[e]; d = ei[E + e]; }
    else       { s = d = e - E; }           // appended self loop
}

// pass 1: segment max of LeakyReLU(a_src[s]+a_dst[d]) over dst (flipped-uint atomicMax)
__global__ void gat_edge_max(const int* __restrict__ ei, int E, int Etot,
                             const float* __restrict__ as_, const float* __restrict__ ad_,
                             unsigned* __restrict__ m, int H)
{
    int e = blockIdx.x * blockDim.x + threadIdx.x;
    if (e >= Etot) return;
    int s, d; edge_endpoints(ei, e, E, s, d);
    for (int h = 0; h < H; ++h) {
        float v = as_[s * H + h] + ad_[d * H + h];
        v = v > 0.f ? v : 0.2f * v;
        atomicMax(&m[d * H + h], flip_f(v));
    }
}

// flipped-uint -> float in place (u==0 means "no edge" -> 0, per reference isfinite guard)
__global__ void gat_unflip(unsigned* __restrict__ m, int total)
{
    int t = blockIdx.x * blockDim.x + threadIdx.x;
    if (t >= total) return;
    unsigned u = m[t];
    ((float*)m)[t] = (u == 0u) ? 0.0f : unflip_f(u);
}

// pass 2: s[d,h] += exp(e - m[d,h])
__global__ void gat_edge_sum(const int* __restrict__ ei, int E, int Etot,
                             const float* __restrict__ as_, const float* __restrict__ ad_,
                             const float* __restrict__ mf, float* __restrict__ sb, int H)
{
    int e = blockIdx.x * blockDim.x + threadIdx.x;
    if (e >= Etot) return;
    int s, d; edge_endpoints(ei, e, E, s, d);
    for (int h = 0; h < H; ++h) {
        float v = as_[s * H + h] + ad_[d * H + h];
        v = v > 0.f ? v : 0.2f * v;
        atomicAdd(&sb[d * H + h], __expf(v - mf[d * H + h]));
    }
}

__global__ void gat_rcp(float* __restrict__ sb, int total)
{
    int t = blockIdx.x * blockDim.x + threadIdx.x;
    if (t >= total) return;
    sb[t] = 1.0f / fmaxf(sb[t], 1e-16f);
}

// pass 3: out[d, h, c] += alpha(e,h) * h1[s, h, c]; one thread per (edge, h*C+c)
// consecutive lanes share an edge -> coalesced atomic bursts on out[dst]
__global__ void gat_edge_agg(const int* __restrict__ ei, int E, int Etot,
                             const float* __restrict__ as_, const float* __restrict__ ad_,
                             const float* __restrict__ mf, const float* __restrict__ rs,
                             const float* __restrict__ hsrc, float* __restrict__ out,
                             int H, int C)
{
    const int HC = H * C;
    long long t = (long long)blockIdx.x * blockDim.x + threadIdx.x;
    long long tot = (long long)Etot * HC;
    if (t >= tot) return;
    int hc = (int)(t % HC);
    int e  = (int)(t / HC);
    int hh = hc / C;
    int s, d; edge_endpoints(ei, e, E, s, d);
    __builtin_prefetch(&hsrc[(size_t)s * HC + hc], 0, 0);
    float v = as_[s * H + hh] + ad_[d * H + hh];
    v = v > 0.f ? v : 0.2f * v;
    float alpha = __expf(v - mf[d * H + hh]) * rs[d * H + hh];
    atomicAdd(&out[(size_t)d * HC + hc], alpha * hsrc[(size_t)s * HC + hc]);
}

// out += bias ; out = ELU(out)
__global__ void gat_bias_elu(float* __restrict__ o, const float* __restrict__ b,
                             long long total, int HC)
{
    long long t = (long long)blockIdx.x * blockDim.x + threadIdx.x;
    if (t >= total) return;
    float v = o[t] + b[t % HC];
    o[t] = v > 0.f ? v : expm1f(v);
}

// in-place: out[n,:] = log_softmax(out[n,:] + bias)
__global__ void gat_logsoftmax(float* __restrict__ out, const float* __restrict__ b,
                               int N, int C)
{
    int n = blockIdx.x * blockDim.x + threadIdx.x;
    if (n >= N) return;
    float* row = out + (size_t)n * C;
    float mx = -INFINITY;
    for (int c = 0; c < C; ++c) mx = fmaxf(mx, row[c] + b[c]);
    float sum = 0.f;
    for (int c = 0; c < C; ++c) sum += __expf(row[c] + b[c] - mx);
    float lse = mx + logf(sum);
    for (int c = 0; c < C; ++c) row[c] = row[c] + b[c] - lse;
}

static inline unsigned cdiv(long long a, long long b) { return (unsigned)((a + b - 1) / b); }

extern "C" void kernel_launch(void* const* d_in, const int* in_sizes, int n_in,
                              void* d_out, int out_size, void* d_ws, size_t ws_size,
                              hipStream_t stream)
{
    const float* x    = (const float*)d_in[0];
    const int*   ei   = (const int*)  d_in[1];
    const float* W1   = (const float*)d_in[2];
    const float* at_s1= (const float*)d_in[3];
    const float* at_d1= (const float*)d_in[4];
    const float* b1   = (const float*)d_in[5];
    const float* W2   = (const float*)d_in[6];
    const float* at_s2= (const float*)d_in[7];
    const float* at_d2= (const float*)d_in[8];
    const float* b2   = (const float*)d_in[9];

    const int F_in = 128;
    const int N    = in_sizes[0] / F_in;         // 100000
    const int E    = in_sizes[1] / 2;            // 1600000
    const int Etot = E + N;                      // + self loops
    const int H1 = 8;
    const int HC1 = in_sizes[5];                 // 64
    const int C1 = HC1 / H1;                     // 8
    const int C2 = in_sizes[9];                  // 40

    // workspace layout (fp32): h1[N*64] | out1[N*64] | aS[N*8] | aD[N*8] | m[N*8] | s[N*8]
    float* h1   = (float*)d_ws;
    float* out1 = h1   + (size_t)N * HC1;
    float* aS   = out1 + (size_t)N * HC1;
    float* aD   = aS   + (size_t)N * H1;
    float* mB   = aD   + (size_t)N * H1;
    float* sB   = mB   + (size_t)N * H1;
    float* outF = (float*)d_out;                 // N*C2

    // ---------------- Layer 1 (H=8, C=8, concat) ----------------
    gat_gemm_wmma<<<N / 16, 128, 0, stream>>>(x, W1, h1, F_in, HC1);
    gat_att_scores<<<cdiv((long long)N * H1, 256), 256, 0, stream>>>(h1, at_s1, at_d1, aS, aD, N, H1, C1);

    // zero: out1 accumulator, flipped-max (0 == most negative), exp-sum
    hipMemsetAsync(out1, 0, (size_t)N * HC1 * sizeof(float), stream);
    hipMemsetAsync(mB,   0, (size_t)N * H1  * sizeof(float), stream);
    hipMemsetAsync(sB,   0, (size_t)N * H1  * sizeof(float), stream);

    gat_edge_max<<<cdiv(Etot, 256), 256, 0, stream>>>(ei, E, Etot, aS, aD, (unsigned*)mB, H1);
    gat_unflip<<<cdiv((long long)N * H1, 256), 256, 0, stream>>>((unsigned*)mB, N * H1);
    gat_edge_sum<<<cdiv(Etot, 256), 256, 0, stream>>>(ei, E, Etot, aS, aD, mB, sB, H1);
    gat_rcp<<<cdiv((long long)N * H1, 256), 256, 0, stream>>>(sB, N * H1);
    gat_edge_agg<<<cdiv((long long)Etot * HC1, 256), 256, 0, stream>>>(ei, E, Etot, aS, aD, mB, sB, h1, out1, H1, C1);
    gat_bias_elu<<<cdiv((long long)N * HC1, 256), 256, 0, stream>>>(out1, b1, (long long)N * HC1, HC1);

    // ---------------- Layer 2 (H=1, C=40, mean == identity) ----------------
    float* h2 = h1;                              // reuse (N*40 <= N*64)
    gat_gemm_wmma<<<N / 16, 128, 0, stream>>>(out1, W2, h2, HC1, C2);
    gat_att_scores<<<cdiv(N, 256), 256, 0, stream>>>(h2, at_s2, at_d2, aS, aD, N, 1, C2);

    hipMemsetAsync(outF, 0, (size_t)N * C2 * sizeof(float), stream);
    hipMemsetAsync(mB,   0, (size_t)N * sizeof(float), stream);
    hipMemsetAsync(sB,   0, (size_t)N * sizeof(float), stream);

    gat_edge_max<<<cdiv(Etot, 256), 256, 0, stream>>>(ei, E, Etot, aS, aD, (unsigned*)mB, 1);
    gat_unflip<<<cdiv(N, 256), 256, 0, stream>>>((unsigned*)mB, N);
    gat_edge_sum<<<cdiv(Etot, 256), 256, 0, stream>>>(ei, E, Etot, aS, aD, mB, sB, 1);
    gat_rcp<<<cdiv(N, 256), 256, 0, stream>>>(sB, N);
    gat_edge_agg<<<cdiv((long long)Etot * C2, 256), 256, 0, stream>>>(ei, E, Etot, aS, aD, mB, sB, h2, outF, 1, C2);

    gat_logsoftmax<<<cdiv(N, 256), 256, 0, stream>>>(outF, b2, N, C2);
}